// BASE_22179211117224
// MI455X (gfx1250) — compile-verified
//
#include <hip/hip_runtime.h>
#include <math.h>

// Problem constants (match reference)
#define CC   256
#define HH   96
#define WW   96
#define HWP  (HH * WW)     // 9216
#define REDN 16

typedef __attribute__((ext_vector_type(2))) float v2f;
typedef __attribute__((ext_vector_type(8))) float v8f;

// ---------------------------------------------------------------------------
// Kernel 1: global average pool over H*W per channel.  grid=256, block=256
// ---------------------------------------------------------------------------
__global__ void se_pool_kernel(const float* __restrict__ x, float* __restrict__ pooled) {
    int c = blockIdx.x;
    int t = threadIdx.x;
    const float* xc = x + (size_t)c * HWP;
    float s = 0.f;
    for (int i = t; i < HWP; i += 256) s += xc[i];
    __shared__ float sm[256];
    sm[t] = s;
    __syncthreads();
    for (int off = 128; off > 0; off >>= 1) {
        if (t < off) sm[t] += sm[t + off];
        __syncthreads();
    }
    if (t == 0) pooled[c] = sm[0] * (1.f / (float)HWP);
}

// ---------------------------------------------------------------------------
// Kernel 2: SE MLP 256 -> 16 (ReLU) -> 256 (sigmoid).  grid=1, block=256
// ---------------------------------------------------------------------------
__global__ void se_mlp_kernel(const float* __restrict__ pooled,
                              const float* __restrict__ w1, const float* __restrict__ b1,
                              const float* __restrict__ w2, const float* __restrict__ b2,
                              float* __restrict__ scale) {
    __shared__ float y1[REDN];
    int t = threadIdx.x;
    if (t < REDN) {
        float acc = b1[t];
        const float* wr = w1 + t * CC;
        for (int c = 0; c < CC; ++c) acc += pooled[c] * wr[c];
        y1[t] = fmaxf(acc, 0.f);
    }
    __syncthreads();
    float acc = b2[t];
    const float* wr = w2 + t * REDN;
    #pragma unroll
    for (int r = 0; r < REDN; ++r) acc += y1[r] * wr[r];
    scale[t] = 1.f / (1.f + expf(-acc));
}

// ---------------------------------------------------------------------------
// Kernel 3: out32 = x * scale[c]; sig = sigmoid(out32).  elementwise
// ---------------------------------------------------------------------------
__global__ void se_apply_kernel(const float* __restrict__ x, const float* __restrict__ scale,
                                float* __restrict__ out32, float* __restrict__ sig) {
    int idx = blockIdx.x * 256 + threadIdx.x;
    if (idx >= CC * HWP) return;
    int c = idx / HWP;
    float v = x[idx] * scale[c];
    out32[idx] = v;
    sig[idx]   = 1.f / (1.f + expf(-v));
}

// ---------------------------------------------------------------------------
// Kernel 4: patch-correlation attention. One 256-thread block per pixel.
// Writes csa_out flat as [p*256 + c] (the reference's .view() then reads this
// same buffer as [c*9216 + p]).
// ---------------------------------------------------------------------------
__global__ void patch_attn_kernel(const float* __restrict__ out32,
                                  const float* __restrict__ sig,
                                  float* __restrict__ csa_out) {
    int p = blockIdx.x;                // pixel
    int i = p / WW, j = p % WW;
    int c = threadIdx.x;               // channel
    __shared__ float red[256];
    __shared__ float avec[9];
    __shared__ float wgt[9];

    float sc = sig[(size_t)c * HWP + p];     // center sigmoid value (1x1 patch)

    #pragma unroll
    for (int k = 0; k < 9; ++k) {
        int di = k / 3 - 1, dj = k % 3 - 1;
        int ii = i + di, jj = j + dj;
        float nb = 0.f;                      // zero-padding AFTER sigmoid
        if (ii >= 0 && ii < HH && jj >= 0 && jj < WW)
            nb = sig[(size_t)c * HWP + ii * WW + jj];
        red[c] = sc * nb;
        __syncthreads();
        for (int off = 128; off > 0; off >>= 1) {
            if (c < off) red[c] += red[c + off];
            __syncthreads();
        }
        if (c == 0) avec[k] = red[0] * (1.f / (float)CC);
        __syncthreads();
    }

    if (c == 0) {                            // softmax over 9 patch positions
        float m = avec[0];
        #pragma unroll
        for (int k = 1; k < 9; ++k) m = fmaxf(m, avec[k]);
        float e[9], s = 0.f;
        #pragma unroll
        for (int k = 0; k < 9; ++k) { e[k] = expf(avec[k] - m); s += e[k]; }
        float inv = 1.f / s;
        #pragma unroll
        for (int k = 0; k < 9; ++k) wgt[k] = e[k] * inv;
    }
    __syncthreads();

    float acc = 0.f;
    #pragma unroll
    for (int k = 0; k < 9; ++k) {
        int di = k / 3 - 1, dj = k % 3 - 1;
        int ii = i + di, jj = j + dj;
        float nb = 0.f;
        if (ii >= 0 && ii < HH && jj >= 0 && jj < WW)
            nb = out32[(size_t)c * HWP + ii * WW + jj];
        acc += wgt[k] * nb;
    }
    csa_out[(size_t)p * CC + c] = acc;       // (h*w, c) layout
}

// ---------------------------------------------------------------------------
// Kernel 5: down GEMM with WMMA f32:  z[o,p] = sum_c dw[o,c] * B[c,p]
//   B[c,p] = csa_out viewed flat as [c*9216 + p]  (the .view() trick)
// One wave computes a 16x16 D tile, K stepped by 4 (V_WMMA_F32_16X16X4_F32).
//   A 16x4 frag: lane L, vgpr v -> A[L%16, (L>>4)*2 + v]
//   B 4x16 frag: lane L, vgpr v -> B[(L>>4)*2 + v, L%16]
//   D 16x16:     vgpr j -> M = j + 8*(L>>4), N = L%16
// grid = (576, 16), block = 32
// ---------------------------------------------------------------------------
__global__ void down_gemm_wmma_kernel(const float* __restrict__ dw,
                                      const float* __restrict__ bmat,
                                      float* __restrict__ z) {
    int tile_n = blockIdx.x * 16;
    int tile_m = blockIdx.y * 16;
    int lane  = threadIdx.x;        // 0..31 (wave32)
    int row   = lane & 15;          // M for A-frag, N for B-frag
    int khalf = (lane >> 4) * 2;    // K base held by this lane

    v8f acc = {};
    const float* arow = dw + (size_t)(tile_m + row) * CC;
    for (int k0 = 0; k0 < CC; k0 += 4) {
        v2f a, b;
        a.x = arow[k0 + khalf + 0];
        a.y = arow[k0 + khalf + 1];
        b.x = bmat[(size_t)(k0 + khalf + 0) * HWP + tile_n + row];
        b.y = bmat[(size_t)(k0 + khalf + 1) * HWP + tile_n + row];
        acc = __builtin_amdgcn_wmma_f32_16x16x4_f32(
            /*neg_a=*/false, a, /*neg_b=*/false, b,
            /*c_mod=*/(short)0, acc, /*reuse_a=*/false, /*reuse_b=*/false);
    }

    int n     = lane & 15;
    int mbase = (lane >> 4) * 8;
    #pragma unroll
    for (int jj = 0; jj < 8; ++jj)
        z[(size_t)(tile_m + mbase + jj) * HWP + tile_n + n] = acc[jj];
}

// ---------------------------------------------------------------------------
// Kernel 6: per-channel mean / rstd for InstanceNorm.  grid=256, block=256
// ---------------------------------------------------------------------------
__global__ void in_stats_kernel(const float* __restrict__ z, float* __restrict__ stats) {
    int o = blockIdx.x;
    int t = threadIdx.x;
    const float* zo = z + (size_t)o * HWP;
    float s = 0.f, s2 = 0.f;
    for (int i = t; i < HWP; i += 256) {
        float v = zo[i];
        s += v; s2 += v * v;
    }
    __shared__ float sm[256], sm2[256];
    sm[t] = s; sm2[t] = s2;
    __syncthreads();
    for (int off = 128; off > 0; off >>= 1) {
        if (t < off) { sm[t] += sm[t + off]; sm2[t] += sm2[t + off]; }
        __syncthreads();
    }
    if (t == 0) {
        float mu  = sm[0]  * (1.f / (float)HWP);
        float var = sm2[0] * (1.f / (float)HWP) - mu * mu;   // biased var
        stats[o * 2 + 0] = mu;
        stats[o * 2 + 1] = rsqrtf(var + 1e-5f);
    }
}

// ---------------------------------------------------------------------------
// Kernel 7: normalize + LeakyReLU(0.2) -> output
// ---------------------------------------------------------------------------
__global__ void in_apply_kernel(const float* __restrict__ z, const float* __restrict__ stats,
                                float* __restrict__ out) {
    int idx = blockIdx.x * 256 + threadIdx.x;
    if (idx >= CC * HWP) return;
    int o = idx / HWP;
    float v = (z[idx] - stats[o * 2]) * stats[o * 2 + 1];
    out[idx] = (v >= 0.f) ? v : 0.2f * v;
}

// ---------------------------------------------------------------------------
// Launch
// ---------------------------------------------------------------------------
extern "C" void kernel_launch(void* const* d_in, const int* in_sizes, int n_in,
                              void* d_out, int out_size, void* d_ws, size_t ws_size,
                              hipStream_t stream) {
    (void)in_sizes; (void)n_in; (void)out_size; (void)ws_size;
    const float* x     = (const float*)d_in[0];
    const float* se_w1 = (const float*)d_in[1];
    const float* se_b1 = (const float*)d_in[2];
    const float* se_w2 = (const float*)d_in[3];
    const float* se_b2 = (const float*)d_in[4];
    const float* dw    = (const float*)d_in[5];
    float* out = (float*)d_out;

    // Workspace layout (floats). N = 256*9216 = 2,359,296.
    // z aliases sig (sig is dead after patch_attn). Peak use: ~28.3 MB.
    const size_t N = (size_t)CC * HWP;
    float* ws      = (float*)d_ws;
    float* pooled  = ws;                 // 256
    float* scale   = ws + 256;           // 256
    float* stats   = ws + 512;           // 512
    float* out32   = ws + 1024;          // N
    float* sig     = out32 + N;          // N
    float* csa_out = sig + N;            // N
    float* z       = sig;                // alias: reuse sig region for z

    int nelem_blocks = (int)((N + 255) / 256);

    se_pool_kernel <<<CC, 256, 0, stream>>>(x, pooled);
    se_mlp_kernel  <<<1, CC, 0, stream>>>(pooled, se_w1, se_b1, se_w2, se_b2, scale);
    se_apply_kernel<<<nelem_blocks, 256, 0, stream>>>(x, scale, out32, sig);
    patch_attn_kernel<<<HWP, CC, 0, stream>>>(out32, sig, csa_out);

    dim3 ggrid(HWP / 16, CC / 16);       // 576 x 16 waves, one per 16x16 tile
    down_gemm_wmma_kernel<<<ggrid, 32, 0, stream>>>(dw, csa_out, z);

    in_stats_kernel<<<CC, 256, 0, stream>>>(z, stats);
    in_apply_kernel<<<nelem_blocks, 256, 0, stream>>>(z, stats, out);
}